// TransformerDist_45947560132767
// MI455X (gfx1250) — compile-verified
//
#include <hip/hip_runtime.h>
#include <math.h>

// Problem constants (from reference)
#define L_   4
#define D_   512
#define H_   8
#define DH_  64
#define KB_  10
#define M_   2048
#define B_   4
#define N_   512
#define BN_  (B_ * N_)      // 2048

typedef __attribute__((ext_vector_type(2))) float v2f;
typedef __attribute__((ext_vector_type(8))) float v8f;

// ---------------------------------------------------------------------------
// Strided batched GEMM, compile-time element strides.
// C[m,n] = act( sum_k A[m,k]*B[k,n] + bias[n] ) (+ resid)
// One wave computes a 64x64 output tile: 4x4 grid of 16x16 WMMA accumulators.
// Per K-step of 4: 4 A-fragments + 4 B-fragments feed 16 v_wmma_f32_16x16x4_f32
// (16 independent accumulation chains; 1 dword load per lane per WMMA).
// Fragment layouts per CDNA5 ISA 7.12.2:
//   A 16x4 f32 : lane = {M = lane&15, Kbase = (lane>>4)*2}, VGPR j -> A[M][Kbase+j]
//   B 4x16 f32 : lane = {N = lane&15, Kbase = (lane>>4)*2}, VGPR j -> B[Kbase+j][N]
//   C/D 16x16  : lane = {N = lane&15, Mbase = (lane>>4)*8}, VGPR j -> C[Mbase+j][N]
// ---------------------------------------------------------------------------
template<int SAR, int SAC, int SBR, int SBC, int SCR, int SCC,
         int ACT, int HAS_BIAS, int HAS_RESID>
__global__ __launch_bounds__(128)
void gemm64_wmma_f32(const float* __restrict__ A, const float* __restrict__ B,
                     const float* __restrict__ bias, const float* __restrict__ resid,
                     float* __restrict__ C,
                     int M, int N, int Kd,
                     long long sAy, long long sAz,
                     long long sBy, long long sBz,
                     long long sCy, long long sCz)
{
    const int lane  = threadIdx.x;            // 0..31
    const int tilesN = N >> 6;
    const int tile  = blockIdx.x * blockDim.y + threadIdx.y;
    if (tile >= (M >> 6) * tilesN) return;    // wave-uniform: EXEC stays all-1s

    const int tm = (tile / tilesN) << 6;
    const int tn = (tile % tilesN) << 6;

    const int l15 = lane & 15;
    const int kh  = (lane >> 4) << 1;         // 0 or 2

    const float* Ap = A + sAy * blockIdx.y + sAz * blockIdx.z
                        + (long long)(tm + l15) * SAR + (long long)kh * SAC;
    const float* Bp = B + sBy * blockIdx.y + sBz * blockIdx.z
                        + (long long)kh * SBR + (long long)(tn + l15) * SBC;

    const v8f vzero = {0.f,0.f,0.f,0.f,0.f,0.f,0.f,0.f};
    v8f acc[4][4];
#pragma unroll
    for (int i = 0; i < 4; ++i)
#pragma unroll
        for (int j = 0; j < 4; ++j) acc[i][j] = vzero;

    for (int k = 0; k < Kd; k += 4) {
        v2f af[4], bf[4];
#pragma unroll
        for (int i = 0; i < 4; ++i) {
            af[i].x = Ap[i * 16 * SAR];
            af[i].y = Ap[i * 16 * SAR + SAC];
            bf[i].x = Bp[i * 16 * SBC];
            bf[i].y = Bp[i * 16 * SBC + SBR];
        }
#pragma unroll
        for (int i = 0; i < 4; ++i)
#pragma unroll
            for (int j = 0; j < 4; ++j)
                acc[i][j] = __builtin_amdgcn_wmma_f32_16x16x4_f32(
                    false, af[i], false, bf[j], (short)0, acc[i][j], false, false);
        Ap += 4 * SAC;
        Bp += 4 * SBR;
    }

    const int mb = (lane >> 4) << 3;          // 0 or 8
    const long long offC = sCy * blockIdx.y + sCz * blockIdx.z;

#pragma unroll
    for (int j = 0; j < 4; ++j) {
        const int cn = tn + j * 16 + l15;
        const float bval = HAS_BIAS ? bias[cn] : 0.0f;
#pragma unroll
        for (int i = 0; i < 4; ++i) {
#pragma unroll
            for (int r = 0; r < 8; ++r) {
                float v = acc[i][j][r] + bval;
                if (ACT) v = 0.5f * v * (1.0f + erff(v * 0.7071067811865475f));
                const long long ci = offC
                    + (long long)(tm + i * 16 + mb + r) * SCR
                    + (long long)cn * SCC;
                if (HAS_RESID) v += resid[ci];
                C[ci] = v;
            }
        }
    }
}

// ---------------------------------------------------------------------------
// LayerNorm over last dim (512). One wave per row.
// ---------------------------------------------------------------------------
__global__ __launch_bounds__(256)
void ln_kernel(const float* __restrict__ x, const float* __restrict__ g,
               const float* __restrict__ b, float* __restrict__ out, int rows)
{
    const int row = blockIdx.x * blockDim.y + threadIdx.y;
    if (row >= rows) return;
    const int lane = threadIdx.x;
    const float* xr = x + (long long)row * D_;

    float vals[16];
    float s = 0.f;
#pragma unroll
    for (int i = 0; i < 16; ++i) {
        float v = xr[lane + 32 * i];
        vals[i] = v; s += v;
    }
#pragma unroll
    for (int off = 16; off > 0; off >>= 1) s += __shfl_xor(s, off, 32);
    const float mean = s * (1.0f / (float)D_);

    float s2 = 0.f;
#pragma unroll
    for (int i = 0; i < 16; ++i) { float d = vals[i] - mean; s2 += d * d; }
#pragma unroll
    for (int off = 16; off > 0; off >>= 1) s2 += __shfl_xor(s2, off, 32);
    const float rstd = rsqrtf(s2 * (1.0f / (float)D_) + 1e-5f);

    float* orow = out + (long long)row * D_;
#pragma unroll
    for (int i = 0; i < 16; ++i) {
        int c = lane + 32 * i;
        orow[c] = (vals[i] - mean) * rstd * g[c] + b[c];
    }
}

// ---------------------------------------------------------------------------
// Distance-bin indices: idx[b,n,m] = clip(floor(|coord[b,n]-coord[b,m]|), 0, 9)
// ---------------------------------------------------------------------------
__global__ __launch_bounds__(256)
void idx_kernel(const float* __restrict__ coord, int* __restrict__ idx)
{
    const int t = blockIdx.x * blockDim.x + threadIdx.x;
    if (t >= B_ * N_ * N_) return;
    const int m = t & (N_ - 1);
    const int n = (t >> 9) & (N_ - 1);
    const int b = t >> 18;
    const float dx = coord[((b * N_ + n) << 1) + 0] - coord[((b * N_ + m) << 1) + 0];
    const float dy = coord[((b * N_ + n) << 1) + 1] - coord[((b * N_ + m) << 1) + 1];
    int i = (int)floorf(sqrtf(dx * dx + dy * dy));
    i = i < 0 ? 0 : (i > KB_ - 1 ? KB_ - 1 : i);
    idx[t] = i;
}

// ---------------------------------------------------------------------------
// p[b,h,n,bin] = sum_d q[b,n,h,d] * rel_k[bin,d]
// ---------------------------------------------------------------------------
__global__ __launch_bounds__(256)
void relscore_kernel(const float* __restrict__ q, const float* __restrict__ relk,
                     float* __restrict__ p)
{
    const int t = blockIdx.x * blockDim.x + threadIdx.x;
    if (t >= B_ * H_ * N_ * KB_) return;
    const int bin = t % KB_;
    const int n   = (t / KB_) & (N_ - 1);
    const int h   = (t / (KB_ * N_)) & (H_ - 1);
    const int b   = t / (KB_ * N_ * H_);
    const float* qp = q + (((long long)(b * N_ + n) * H_ + h) << 6);
    const float* rp = relk + bin * DH_;
    float s = 0.f;
#pragma unroll 8
    for (int d = 0; d < DH_; ++d) s += qp[d] * rp[d];
    p[t] = s;
}

// ---------------------------------------------------------------------------
// Fused: scores = (S + p[idx]) * scale -> softmax row -> write attn back,
// and per-bin attention mass w[b,h,n,bin]. One wave per (b,h,n) row.
// ---------------------------------------------------------------------------
__global__ __launch_bounds__(256)
void softmax_kernel(float* __restrict__ S, const int* __restrict__ idx,
                    const float* __restrict__ p, float* __restrict__ w)
{
    const int row = blockIdx.x * blockDim.y + threadIdx.y;   // 0..B*H*N-1
    if (row >= B_ * H_ * N_) return;
    const int lane = threadIdx.x;
    const int n = row & (N_ - 1);
    const int b = row >> 12;                                  // row / (H_*N_)

    float* Sr = S + (long long)row * N_;
    const int* ir = idx + (long long)(b * N_ + n) * N_;
    const float* pr = p + (long long)row * KB_;
    const float scale = 0.125f;                               // DH^-0.5

    float sc[16];
    int   bins[16];
    float mx = -3.0e38f;
#pragma unroll
    for (int i = 0; i < 16; ++i) {
        int c = lane + 32 * i;
        int bn = ir[c];
        float v = (Sr[c] + pr[bn]) * scale;
        sc[i] = v; bins[i] = bn;
        mx = fmaxf(mx, v);
    }
#pragma unroll
    for (int off = 16; off > 0; off >>= 1) mx = fmaxf(mx, __shfl_xor(mx, off, 32));

    float sum = 0.f;
#pragma unroll
    for (int i = 0; i < 16; ++i) { sc[i] = expf(sc[i] - mx); sum += sc[i]; }
#pragma unroll
    for (int off = 16; off > 0; off >>= 1) sum += __shfl_xor(sum, off, 32);
    const float inv = 1.0f / sum;

#pragma unroll
    for (int i = 0; i < 16; ++i) {
        sc[i] *= inv;
        Sr[lane + 32 * i] = sc[i];
    }
    // per-bin attention mass
    for (int k = 0; k < KB_; ++k) {
        float acc = 0.f;
#pragma unroll
        for (int i = 0; i < 16; ++i) acc += (bins[i] == k) ? sc[i] : 0.f;
#pragma unroll
        for (int off = 16; off > 0; off >>= 1) acc += __shfl_xor(acc, off, 32);
        if (lane == 0) w[(long long)row * KB_ + k] = acc;
    }
}

// ---------------------------------------------------------------------------
// o[b,n,h,d] += sum_bin w[b,h,n,bin] * rel_v[bin,d]
// ---------------------------------------------------------------------------
__global__ __launch_bounds__(256)
void relvadd_kernel(float* __restrict__ o, const float* __restrict__ w,
                    const float* __restrict__ relv)
{
    const int t = blockIdx.x * blockDim.x + threadIdx.x;
    if (t >= BN_ * D_) return;
    const int d = t & (DH_ - 1);
    const int h = (t >> 6) & (H_ - 1);
    const int n = (t >> 9) & (N_ - 1);
    const int b = t >> 18;
    const float* wr = w + (long long)((b * H_ + h) * N_ + n) * KB_;
    float acc = o[t];
#pragma unroll
    for (int k = 0; k < KB_; ++k) acc += wr[k] * relv[k * DH_ + d];
    o[t] = acc;
}

__global__ __launch_bounds__(256)
void copy_kernel(const float* __restrict__ src, float* __restrict__ dst, int n)
{
    const int t = blockIdx.x * blockDim.x + threadIdx.x;
    if (t < n) dst[t] = src[t];
}

// ---------------------------------------------------------------------------
// Host-side launch helper
// ---------------------------------------------------------------------------
template<int SAR, int SAC, int SBR, int SBC, int SCR, int SCC,
         int ACT, int HAS_BIAS, int HAS_RESID>
static inline void gemm_go(hipStream_t stream,
                           const float* A, const float* B, const float* bias,
                           const float* resid, float* C,
                           int M, int N, int Kd,
                           long long sAy, long long sAz,
                           long long sBy, long long sBz,
                           long long sCy, long long sCz,
                           int gy, int gz)
{
    const int tiles = (M >> 6) * (N >> 6);
    dim3 grid((tiles + 3) / 4, gy, gz);
    dim3 blk(32, 4);
    gemm64_wmma_f32<SAR, SAC, SBR, SBC, SCR, SCC, ACT, HAS_BIAS, HAS_RESID>
        <<<grid, blk, 0, stream>>>(A, B, bias, resid, C, M, N, Kd,
                                   sAy, sAz, sBy, sBz, sCy, sCz);
}

extern "C" void kernel_launch(void* const* d_in, const int* in_sizes, int n_in,
                              void* d_out, int out_size, void* d_ws, size_t ws_size,
                              hipStream_t stream)
{
    const float* x     = (const float*)d_in[0];
    const float* coord = (const float*)d_in[1];
    const float* ln1_g = (const float*)d_in[2];
    const float* ln1_b = (const float*)d_in[3];
    const float* wq = (const float*)d_in[4];
    const float* bq = (const float*)d_in[5];
    const float* wk = (const float*)d_in[6];
    const float* bk = (const float*)d_in[7];
    const float* wv = (const float*)d_in[8];
    const float* bv = (const float*)d_in[9];
    const float* wo = (const float*)d_in[10];
    const float* bo = (const float*)d_in[11];
    const float* rel_k = (const float*)d_in[12];
    const float* rel_v = (const float*)d_in[13];
    const float* ln2_g = (const float*)d_in[14];
    const float* ln2_b = (const float*)d_in[15];
    const float* w1 = (const float*)d_in[16];
    const float* b1 = (const float*)d_in[17];
    const float* w2 = (const float*)d_in[18];
    const float* b2 = (const float*)d_in[19];
    const float* lnf_g = (const float*)d_in[20];
    const float* lnf_b = (const float*)d_in[21];
    float* out = (float*)d_out;

    // workspace carve-up (floats)
    float* xb = (float*)d_ws;                    // [2048,512] running residual
    float* hb = xb + BN_ * D_;                   // [2048,512] LN output
    float* qb = hb + BN_ * D_;                   // [B,N,H,DH]
    float* kb = qb + BN_ * D_;
    float* vb = kb + BN_ * D_;
    float* ob = vb + BN_ * D_;                   // attention output pre-proj
    float* gb = ob + BN_ * D_;                   // [2048,2048] FFN hidden
    float* Sb = gb + (size_t)BN_ * M_;           // [B,H,N,N] scores/attn (32MB)
    float* pb = Sb + (size_t)B_ * H_ * N_ * N_;  // [B,H,N,KB]
    float* wb2 = pb + B_ * H_ * N_ * KB_;        // [B,H,N,KB]
    int*   idxb = (int*)(wb2 + B_ * H_ * N_ * KB_); // [B,N,N]

    copy_kernel<<<(BN_ * D_ + 255) / 256, 256, 0, stream>>>(x, xb, BN_ * D_);
    idx_kernel<<<(B_ * N_ * N_ + 255) / 256, 256, 0, stream>>>(coord, idxb);

    const long long NBH = (long long)N_ * D_;            // per-b stride in q/k/v
    const long long SBH = (long long)N_ * N_;            // per-h stride in S
    const long long SBB = (long long)H_ * N_ * N_;       // per-b stride in S

    for (int l = 0; l < L_; ++l) {
        const float* wq_l = wq + (long long)l * D_ * D_;
        const float* wk_l = wk + (long long)l * D_ * D_;
        const float* wv_l = wv + (long long)l * D_ * D_;
        const float* wo_l = wo + (long long)l * D_ * D_;
        const float* w1_l = w1 + (long long)l * D_ * M_;
        const float* w2_l = w2 + (long long)l * M_ * D_;

        // h = LN1(x)
        ln_kernel<<<BN_ / 8, dim3(32, 8), 0, stream>>>(xb, ln1_g + l * D_, ln1_b + l * D_, hb, BN_);

        // q,k,v = h @ W + b   ([2048,512] x [512,512])
        gemm_go<D_, 1, D_, 1, D_, 1, 0, 1, 0>(stream, hb, wq_l, bq + l * D_, nullptr, qb,
                BN_, D_, D_, 0, 0, 0, 0, 0, 0, 1, 1);
        gemm_go<D_, 1, D_, 1, D_, 1, 0, 1, 0>(stream, hb, wk_l, bk + l * D_, nullptr, kb,
                BN_, D_, D_, 0, 0, 0, 0, 0, 0, 1, 1);
        gemm_go<D_, 1, D_, 1, D_, 1, 0, 1, 0>(stream, hb, wv_l, bv + l * D_, nullptr, vb,
                BN_, D_, D_, 0, 0, 0, 0, 0, 0, 1, 1);

        // p[b,h,n,bin] = q . rel_k[bin]
        relscore_kernel<<<(B_ * H_ * N_ * KB_ + 255) / 256, 256, 0, stream>>>(
            qb, rel_k + (long long)l * KB_ * DH_, pb);

        // S[b,h] = q[b,h] @ k[b,h]^T      (M=N=512, K=64; grid y=H, z=B)
        gemm_go<D_, 1, 1, D_, N_, 1, 0, 0, 0>(stream, qb, kb, nullptr, nullptr, Sb,
                N_, N_, DH_,
                DH_, NBH,          // A = q[b,n,h,d]
                DH_, NBH,          // B[k][m] = k[b,m,h,k]
                SBH, SBB,          // C = S[b,h,n,m]
                H_, B_);

        // softmax + relation bias + bin masses
        softmax_kernel<<<B_ * H_ * N_ / 8, dim3(32, 8), 0, stream>>>(Sb, idxb, pb, wb2);

        // o[b,n,h,:] = attn[b,h] @ v[b,h]   (M=512, N=64, K=512)
        gemm_go<N_, 1, D_, 1, D_, 1, 0, 0, 0>(stream, Sb, vb, nullptr, nullptr, ob,
                N_, DH_, N_,
                SBH, SBB,          // A = attn[b,h,n,m]
                DH_, NBH,          // B[m][d] = v[b,m,h,d]
                DH_, NBH,          // C = o[b,n,h,d]
                H_, B_);

        // o += w @ rel_v
        relvadd_kernel<<<(BN_ * D_ + 255) / 256, 256, 0, stream>>>(
            ob, wb2, rel_v + (long long)l * KB_ * DH_);

        // x = o @ wo + bo + x
        gemm_go<D_, 1, D_, 1, D_, 1, 0, 1, 1>(stream, ob, wo_l, bo + l * D_, xb, xb,
                BN_, D_, D_, 0, 0, 0, 0, 0, 0, 1, 1);

        // h = LN2(x)
        ln_kernel<<<BN_ / 8, dim3(32, 8), 0, stream>>>(xb, ln2_g + l * D_, ln2_b + l * D_, hb, BN_);

        // g = gelu(h @ w1 + b1)   ([2048,512] x [512,2048])
        gemm_go<D_, 1, M_, 1, M_, 1, 1, 1, 0>(stream, hb, w1_l, b1 + l * M_, nullptr, gb,
                BN_, M_, D_, 0, 0, 0, 0, 0, 0, 1, 1);

        // x = g @ w2 + b2 + x     ([2048,2048] x [2048,512])
        gemm_go<M_, 1, D_, 1, D_, 1, 0, 1, 1>(stream, gb, w2_l, b2 + l * D_, xb, xb,
                BN_, D_, M_, 0, 0, 0, 0, 0, 0, 1, 1);
    }

    // out = LNf(x)
    ln_kernel<<<BN_ / 8, dim3(32, 8), 0, stream>>>(xb, lnf_g, lnf_b, out, BN_);
}